// OrthoLayer_63745904607766
// MI455X (gfx1250) — compile-verified
//
#include <hip/hip_runtime.h>
#include <cstddef>

typedef __attribute__((ext_vector_type(2))) float v2f;
typedef __attribute__((ext_vector_type(4))) float v4f;
typedef __attribute__((ext_vector_type(8))) float v8f;

#define FDIM 128
#define N_STAGES 253

// ---------------------------------------------------------------------------
// Kernel 1: build Q (128x128) from the Givens cascade.
// One wave per row of Q (rows are independent: rotations act on columns).
// Row lives in LDS; per-wave LDS ops are in-order so no barriers are needed
// between stages. 16 blocks x 8 waves = 128 rows, all parallel.
// ---------------------------------------------------------------------------
__global__ __launch_bounds__(256) void build_q_kernel(
    const float* __restrict__ thetas, float* __restrict__ Qg) {
  __shared__ float rows[8 * FDIM];
  const int lane = threadIdx.x & 31;
  const int wv   = threadIdx.x >> 5;
  const int r    = blockIdx.x * 8 + wv;   // this wave's row of Q
  float* rw = &rows[wv * FDIM];

  // identity row
  #pragma unroll
  for (int i = 0; i < 4; ++i) {
    const int c = lane * 4 + i;
    rw[c] = (c == r) ? 1.0f : 0.0f;
  }
  __builtin_amdgcn_wave_barrier();

  int t0 = 0;
  for (int st = 0; st < N_STAGES; ++st) {
    const int s = st & 1;
    const int e = (st < 127) ? (st + 2) : (254 - st);
    const int n = (e - s) >> 1;

    int j = lane;
    if (j < n) {
      const float th = thetas[t0 + j];
      const float c  = __cosf(th);
      const float si = __sinf(th);
      const int c0 = s + 2 * j;
      const float a = rw[c0];
      const float b = rw[c0 + 1];
      rw[c0]     = a * c - b * si;
      rw[c0 + 1] = a * si + b * c;
    }
    j = lane + 32;
    if (j < n) {
      const float th = thetas[t0 + j];
      const float c  = __cosf(th);
      const float si = __sinf(th);
      const int c0 = s + 2 * j;
      const float a = rw[c0];
      const float b = rw[c0 + 1];
      rw[c0]     = a * c - b * si;
      rw[c0 + 1] = a * si + b * c;
    }
    t0 += n;
    __builtin_amdgcn_wave_barrier();  // keep LDS store->load order across stages
  }

  // write row to global (coalesced b128 per lane)
  const int c = lane * 4;
  v4f out;
  out.x = rw[c + 0];
  out.y = rw[c + 1];
  out.z = rw[c + 2];
  out.w = rw[c + 3];
  *(v4f*)(&Qg[(size_t)r * FDIM + c]) = out;
}

// ---------------------------------------------------------------------------
// Kernel 2: Y = X @ Q + bias via V_WMMA_F32_16X16X4_F32 (full f32 precision).
// Block = 256 threads = 8 waves; each wave computes a 16(M) x 128(N) tile,
// block covers 128 rows. Q staged in LDS, pair-swizzled for bank-conflict-free
// b64 B-loads (odd K-pairs rotated by 32 dwords -> half-waves hit disjoint
// bank halves).
//
// f32 WMMA operand layout (wave32):
//   A 16x4 : lanes 0-15 rows M, {K0,K1}; lanes 16-31 rows M, {K2,K3}
//   B 4x16 : lanes 0-15 cols N, {K0,K1}; lanes 16-31 cols N, {K2,K3}
//   C 16x16: VGPR j = row (j | j+8 for hi lanes), lane&15 = col
// ---------------------------------------------------------------------------
__global__ __launch_bounds__(256) void ortho_gemm_kernel(
    const float* __restrict__ X, const float* __restrict__ Qg,
    const float* __restrict__ bias, float* __restrict__ Y) {
  __shared__ float QL[FDIM * FDIM];  // 64 KB, swizzled

  const int tid = threadIdx.x;
  // Stage Q: element (k,n) -> pair p=k>>1, block of 256 dwords,
  // offset (2n + 32*(p&1)) % 256, +(k&1).
  for (int i = tid; i < FDIM * FDIM; i += 256) {
    const int k = i >> 7;
    const int n = i & 127;
    const int p = k >> 1;
    const int off = (2 * n + ((p & 1) << 5)) & 255;
    QL[p * 256 + off + (k & 1)] = Qg[i];
  }
  __syncthreads();

  const int lane = tid & 31;
  const int wv   = tid >> 5;
  const int lo   = lane & 15;
  const int hi   = lane >> 4;
  const int m0   = blockIdx.x * 128 + wv * 16;

  // per-lane A source: row m0+lo, K offset 2*hi within each chunk of 4
  const float* px = X + (size_t)(m0 + lo) * FDIM + 2 * hi;

  // accumulators pre-loaded with bias (bias is per-column = per-lane constant)
  v8f acc[8];
  #pragma unroll
  for (int nt = 0; nt < 8; ++nt) {
    const float bv = bias[nt * 16 + lo];
    #pragma unroll
    for (int j = 0; j < 8; ++j) acc[nt][j] = bv;
  }

  const int rot = hi << 5;  // odd-pair swizzle rotation for hi half-wave
  #pragma unroll 4
  for (int kc = 0; kc < 32; ++kc) {
    const v2f A = *(const v2f*)(px + kc * 4);           // 8B aligned
    const int pbase = (2 * kc + hi) * 256;
    #pragma unroll
    for (int nt = 0; nt < 8; ++nt) {
      const int off = ((nt * 32 + 2 * lo + rot) & 255);
      const v2f B = *(const v2f*)(&QL[pbase + off]);    // conflict-free b64
      acc[nt] = __builtin_amdgcn_wmma_f32_16x16x4_f32(
          false, A, false, B, (short)0, acc[nt], false, false);
    }
  }

  // store: per (nt,j) the wave writes two contiguous 64B segments
  const int rbase = m0 + hi * 8;
  #pragma unroll
  for (int nt = 0; nt < 8; ++nt) {
    const int col = nt * 16 + lo;
    #pragma unroll
    for (int j = 0; j < 8; ++j) {
      Y[(size_t)(rbase + j) * FDIM + col] = acc[nt][j];
    }
  }
}

extern "C" void kernel_launch(void* const* d_in, const int* in_sizes, int n_in,
                              void* d_out, int out_size, void* d_ws, size_t ws_size,
                              hipStream_t stream) {
  const float* x      = (const float*)d_in[0];
  const float* thetas = (const float*)d_in[1];
  const float* bias   = (const float*)d_in[2];
  float* y  = (float*)d_out;
  float* Qg = (float*)d_ws;  // 128*128 f32 = 64 KB scratch

  const int batch   = in_sizes[0] / FDIM;      // 131072
  const int mblocks = batch / 128;             // 1024

  build_q_kernel<<<16, 256, 0, stream>>>(thetas, Qg);
  ortho_gemm_kernel<<<mblocks, 256, 0, stream>>>(x, Qg, bias, y);
}